// VectorQuantizer_61692910240151
// MI455X (gfx1250) — compile-verified
//
#include <hip/hip_runtime.h>
#include <hip/hip_bf16.h>
#include <math.h>
#include <stdint.h>

typedef float v2f __attribute__((ext_vector_type(2)));
typedef float v8f __attribute__((ext_vector_type(8)));

#define NUM_CODES 8192
#define LATENT    256
#define N_VEC     8192      // 8 * 32 * 32
#define SPATIAL   1024      // 32 * 32
#define BETA      0.25f
#define EPS_NORM  1e-12f
#define TOTAL_ELEMS 2097152.0f   // 8*256*32*32

// LDS row stride in floats: 260*4 bytes is 16B aligned (b128 staging stores ok)
// and 260 mod 64 == 4 -> WMMA-pattern float2 reads are bank-conflict free.
#define PAD        260
#define BLOCK_ROWS 64
#define K2_THREADS 128

// ---------------------------------------------------------------------------
// CDNA5 async copy helpers (ASYNCcnt-tracked, no VGPR round trip).
// LDS destination address = low 32 bits of the generic shared pointer.
// ---------------------------------------------------------------------------
__device__ __forceinline__ void async_load_b128(unsigned lds_off, const void* g) {
    asm volatile("global_load_async_to_lds_b128 %0, %1, off"
                 :: "v"(lds_off), "v"((unsigned long long)(uintptr_t)g)
                 : "memory");
}
__device__ __forceinline__ void wait_async0() {
    asm volatile("s_wait_asynccnt 0x0" ::: "memory");
}

// ---------------------------------------------------------------------------
// Kernel 0: zero histogram + loss accumulator (runs every launch; deterministic)
// ---------------------------------------------------------------------------
__global__ void vq_init_kernel(unsigned int* __restrict__ counts,
                               float* __restrict__ lossacc) {
    int i = blockIdx.x * blockDim.x + threadIdx.x;
    if (i < NUM_CODES) counts[i] = 0u;
    if (i == 0) *lossacc = 0.0f;
}

// ---------------------------------------------------------------------------
// Kernel 1: normalize codebook rows (F.normalize, dim=1) + ||w||^2 per row.
// One wave per row: 32 lanes x 8 floats.
// ---------------------------------------------------------------------------
__global__ __launch_bounds__(256) void vq_normalize_kernel(
        const float* __restrict__ cb,
        float* __restrict__ w,
        float* __restrict__ wnorm) {
    const int lane = threadIdx.x & 31;
    const int wave = threadIdx.x >> 5;
    const int row  = blockIdx.x * 8 + wave;

    const float* src = cb + (size_t)row * LATENT + lane * 8;
    float4 a = *(const float4*)(src);
    float4 b = *(const float4*)(src + 4);

    float ss = a.x * a.x + a.y * a.y + a.z * a.z + a.w * a.w +
               b.x * b.x + b.y * b.y + b.z * b.z + b.w * b.w;
    #pragma unroll
    for (int off = 16; off >= 1; off >>= 1)
        ss += __shfl_xor(ss, off, 32);

    float norm = sqrtf(ss);
    float inv  = 1.0f / fmaxf(norm, EPS_NORM);

    float* dst = w + (size_t)row * LATENT + lane * 8;
    float4 oa = {a.x * inv, a.y * inv, a.z * inv, a.w * inv};
    float4 ob = {b.x * inv, b.y * inv, b.z * inv, b.w * inv};
    *(float4*)(dst)     = oa;
    *(float4*)(dst + 4) = ob;

    if (lane == 0) wnorm[row] = ss * inv * inv;
}

// ---------------------------------------------------------------------------
// Kernel 2: distance argmin via V_WMMA_F32_16X16X4_F32.
// 128 threads = 4 waves; block owns 64 z-rows (staged to LDS once).
// Double-buffered async B staging (global_load_async_to_lds_b128) overlaps
// the next 16-code tile's copy with the current tile's 64 WMMAs, which run
// as two independent accumulator chains to break the WMMA RAW dependency.
// Score = ||w_n||^2 - 2 * z.w  (row norm of z is constant -> dropped).
// ---------------------------------------------------------------------------
__global__ __launch_bounds__(K2_THREADS) void vq_argmin_kernel(
        const float* __restrict__ z,
        const float* __restrict__ w,
        const float* __restrict__ wnorm,
        int* __restrict__ idxbuf) {
    __shared__ float As[BLOCK_ROWS * PAD];   // 64 x 256 z-tile (padded)
    __shared__ float Bs[2][16 * PAD];        // double-buffered 16 x 256 code-tile

    const int tid  = threadIdx.x;
    const int lane = tid & 31;
    const int wave = tid >> 5;
    const int half = lane >> 4;              // 0: K+0..1 / M 0..7 ; 1: K+2..3 / M 8..15
    const int mr   = lane & 15;

    const int rowBase = blockIdx.x * BLOCK_ROWS;
    const int batch   = rowBase >> 10;       // 64 | 1024 -> whole block in one image
    const int sBase   = rowBase & (SPATIAL - 1);

    // Stage A: As[r][d] = z[batch][d][spatial sBase+r]  (NCHW source)
    for (int e = tid; e < BLOCK_ROWS * LATENT; e += K2_THREADS) {
        int r = e & 63;
        int d = e >> 6;
        As[r * PAD + d] = z[((size_t)(batch * LATENT + d)) * SPATIAL + sBase + r];
    }

    // Async stage of one 16x256 B tile into Bs[buf]: 8 x b128 per thread.
    auto stageB = [&](int ct, int buf) {
        const float4* wsrc = (const float4*)&w[(size_t)ct * 16 * LATENT];
        #pragma unroll
        for (int i = 0; i < (16 * LATENT / 4) / K2_THREADS; ++i) {
            int e4 = tid + i * K2_THREADS;
            int rr = e4 >> 6;                 // 64 float4 per row
            int cc = (e4 & 63) << 2;
            unsigned lds_off =
                (unsigned)(uintptr_t)(const void*)&Bs[buf][rr * PAD + cc];
            async_load_b128(lds_off, (const void*)(wsrc + e4));
        }
    };

    stageB(0, 0);                            // prologue: tile 0 in flight
    __syncthreads();                         // A tile visible to all waves

    // Per-lane pointers matching the 16x4 f32 A VGPR layout.
    const float* Arow = &As[(wave * 16 + mr) * PAD + 2 * half];

    float best[8];
    int   bidx[8];
    #pragma unroll
    for (int v = 0; v < 8; ++v) { best[v] = 3.4e38f; bidx[v] = 0; }

    for (int ct = 0; ct < NUM_CODES / 16; ++ct) {
        const int p = ct & 1;
        wait_async0();                       // our async writes for tile ct landed
        __syncthreads();                     // ...and everyone else's too;
                                             // also: all reads of Bs[p^1] retired
        if (ct + 1 < NUM_CODES / 16)
            stageB(ct + 1, p ^ 1);           // overlap next copy with compute

        const float* Brow = &Bs[p][mr * PAD + 2 * half];

        v8f c0 = {0.f, 0.f, 0.f, 0.f, 0.f, 0.f, 0.f, 0.f};
        v8f c1 = {0.f, 0.f, 0.f, 0.f, 0.f, 0.f, 0.f, 0.f};
        #pragma unroll
        for (int kk = 0; kk < LATENT / 4; kk += 2) {
            v2f a0 = *(const v2f*)(Arow + 4 * kk);
            v2f b0 = *(const v2f*)(Brow + 4 * kk);
            c0 = __builtin_amdgcn_wmma_f32_16x16x4_f32(
                    false, a0, false, b0, (short)0, c0, false, false);
            v2f a1 = *(const v2f*)(Arow + 4 * kk + 4);
            v2f b1 = *(const v2f*)(Brow + 4 * kk + 4);
            c1 = __builtin_amdgcn_wmma_f32_16x16x4_f32(
                    false, a1, false, b1, (short)0, c1, false, false);
        }

        const int   code = ct * 16 + mr;
        const float wn   = wnorm[code];
        #pragma unroll
        for (int v = 0; v < 8; ++v) {
            float s = wn - 2.0f * (c0[v] + c1[v]);   // strict < keeps first index
            if (s < best[v]) { best[v] = s; bidx[v] = code; }
        }
    }

    // Cross-lane argmin within each 16-lane half (lanes of one C-matrix row).
    #pragma unroll
    for (int v = 0; v < 8; ++v) {
        float bv = best[v];
        int   iv = bidx[v];
        #pragma unroll
        for (int off = 8; off >= 1; off >>= 1) {
            float ob = __shfl_xor(bv, off, 16);
            int   oi = __shfl_xor(iv, off, 16);
            if (ob < bv || (ob == bv && oi < iv)) { bv = ob; iv = oi; }
        }
        if (mr == 0) {
            int row = rowBase + wave * 16 + v + 8 * half;  // C layout: M = v + 8*half
            idxbuf[row] = iv;
        }
    }
}

// ---------------------------------------------------------------------------
// Kernel 3: gather z_q = w[idx], write NCHW output (straight-through value),
// fused loss accumulation and code-usage histogram. One block per vector n.
// ---------------------------------------------------------------------------
__global__ __launch_bounds__(256) void vq_gather_kernel(
        const float* __restrict__ z,
        const float* __restrict__ w,
        const int* __restrict__ idxbuf,
        float* __restrict__ out_zq,
        float* __restrict__ out_idx,
        float* __restrict__ lossacc,
        unsigned int* __restrict__ counts) {
    __shared__ float red[256];
    const int n = blockIdx.x;
    const int d = threadIdx.x;
    const int code = idxbuf[n];
    const int batch = n >> 10;
    const int s = n & (SPATIAL - 1);

    const size_t gi = ((size_t)(batch * LATENT + d)) * SPATIAL + s;
    const float wv = w[(size_t)code * LATENT + d];
    const float zv = z[gi];
    out_zq[gi] = wv;

    const float diff = wv - zv;
    red[d] = diff * diff;
    __syncthreads();
    #pragma unroll
    for (int off = 128; off >= 1; off >>= 1) {
        if (d < off) red[d] += red[d + off];
        __syncthreads();
    }
    if (d == 0) {
        atomicAdd(lossacc, red[0]);
        atomicAdd(&counts[code], 1u);
        out_idx[n] = (float)code;
    }
}

// ---------------------------------------------------------------------------
// Kernel 4: finalize loss scalar and perplexity from histogram.
// ---------------------------------------------------------------------------
__global__ __launch_bounds__(256) void vq_finalize_kernel(
        const unsigned int* __restrict__ counts,
        const float* __restrict__ lossacc,
        float* __restrict__ out_loss,
        float* __restrict__ out_perp) {
    __shared__ float red[256];
    const int t = threadIdx.x;
    float s = 0.0f;
    for (int i = t; i < NUM_CODES; i += 256) {
        float p = (float)counts[i] * (1.0f / (float)N_VEC);
        s += p * logf(p + 1e-10f);
    }
    red[t] = s;
    __syncthreads();
    #pragma unroll
    for (int off = 128; off >= 1; off >>= 1) {
        if (t < off) red[t] += red[t + off];
        __syncthreads();
    }
    if (t == 0) {
        *out_loss = (1.0f + BETA) * (*lossacc) * (1.0f / TOTAL_ELEMS);
        *out_perp = expf(-red[0]);
    }
}

// ---------------------------------------------------------------------------
extern "C" void kernel_launch(void* const* d_in, const int* in_sizes, int n_in,
                              void* d_out, int out_size, void* d_ws, size_t ws_size,
                              hipStream_t stream) {
    const float* z  = (const float*)d_in[0];   // [8, 256, 32, 32]
    const float* cb = (const float*)d_in[1];   // [8192, 256]

    // Workspace layout (~8.1 MB)
    float* w      = (float*)d_ws;                          // 8192*256 normalized codebook
    float* wnorm  = w + (size_t)NUM_CODES * LATENT;        // 8192
    int*   idxbuf = (int*)(wnorm + NUM_CODES);             // 8192
    unsigned int* counts = (unsigned int*)(idxbuf + NUM_CODES); // 8192
    float* lossacc = (float*)(counts + NUM_CODES);         // 1

    // Output layout: z_q [2097152] | idx-as-float [8192] | loss | perplexity
    float* out      = (float*)d_out;
    float* out_zq   = out;
    float* out_idx  = out + (size_t)N_VEC * LATENT;
    float* out_loss = out_idx + N_VEC;
    float* out_perp = out_loss + 1;

    vq_init_kernel<<<(NUM_CODES + 255) / 256, 256, 0, stream>>>(counts, lossacc);
    vq_normalize_kernel<<<NUM_CODES / 8, 256, 0, stream>>>(cb, w, wnorm);
    vq_argmin_kernel<<<N_VEC / BLOCK_ROWS, K2_THREADS, 0, stream>>>(z, w, wnorm, idxbuf);
    vq_gather_kernel<<<N_VEC, 256, 0, stream>>>(z, w, idxbuf, out_zq, out_idx,
                                                lossacc, counts);
    vq_finalize_kernel<<<1, 256, 0, stream>>>(counts, lossacc, out_loss, out_perp);
}